// SSIM_781684048112
// MI455X (gfx1250) — compile-verified
//
#include <hip/hip_runtime.h>
#include <cmath>

typedef __attribute__((ext_vector_type(2))) float v2f;
typedef __attribute__((ext_vector_type(8))) float v8f;

struct SsimConsts {
    float k[13];     // gaussian taps, sigma=1.5, radius=6
    float M[3][3];   // channel-axis blur collapsed to 3x3 mix (reflect, n=3)
};

#define NIMG 32
#define NCH  3
#define HW   512
#define TILES 32   // 512/16

// One block = one (n, 16x16 spatial tile). 3 waves.
// Phase A: wave w blurs the 5 RAW fields of input channel w (H then W) via
//          banded-matrix WMMA f32 16x16x4, results staged to LDS in D-layout.
// Phase B: wave w applies the 3x3 channel-mix for OUTPUT channel w pointwise
//          on the blurred 16x16 results, computes SSIM, reduces.
__global__ __launch_bounds__(96) void ssim_tile_kernel(
    const float* __restrict__ img1, const float* __restrict__ img2,
    float* __restrict__ partials, SsimConsts sc)
{
    __shared__ float sP1[3][28][29];   // img1 patches (c, row, col), padded
    __shared__ float sP2[3][28][29];   // img2 patches
    __shared__ float sT[3][16][33];    // per-wave H-blurred staging (D->A relayout)
    __shared__ float sOut[3][5][256];  // blurred raw fields, D-layout: [cc][f][lane*8+v]
    __shared__ float sKpad[48];        // taps padded with zeros: sKpad[d+15] = k[d]
    __shared__ float sRed[3];

    const int tid = threadIdx.x;
    const int b   = blockIdx.x;
    const int n   = b >> 10;           // 1024 tiles per image
    const int rem = b & 1023;
    const int th  = rem >> 5;
    const int tw  = rem & 31;

    if (tid < 48) sKpad[tid] = (tid >= 15 && tid <= 27) ? sc.k[tid - 15] : 0.0f;

    // ---- load 28x28 reflect-padded patches for 3 channels of both images ----
    const int row0 = th * 16 - 6;
    const int col0 = tw * 16 - 6;
    for (int idx = tid; idx < 3 * 28 * 28; idx += 96) {
        int cc = idx / 784;
        int r  = (idx % 784) / 28;
        int q  = idx % 28;
        int gi = row0 + r;  int gj = col0 + q;
        int jr = gi & 1023; jr = (jr < HW) ? jr : (1023 - jr);   // reflect (half-sample)
        int jc = gj & 1023; jc = (jc < HW) ? jc : (1023 - jc);
        size_t base = ((size_t)(n * NCH + cc) * HW + jr) * HW + jc;
        sP1[cc][r][q] = img1[base];
        sP2[cc][r][q] = img2[base];
    }
    __syncthreads();

    const int lane = tid & 31;
    const int c    = tid >> 5;       // wave id: input channel (A) / output channel (B)
    const int half = lane >> 4;      // 0: lanes 0-15, 1: lanes 16-31
    const int l15  = lane & 15;
    const int u0   = 2 * half;       // K-pair selector per WMMA 32-bit A/B layout

    auto kb = [&](int d) -> float { return sKpad[d + 15]; };  // k[d] or 0

    // Banded coefficient operand (same values serve pass-1 A and pass-2 B):
    // elem(row=l15, K=4kk+u0{+1}) = k[K - l15]
    v2f kband[7];
#pragma unroll
    for (int kk = 0; kk < 7; ++kk) {
        int K0 = 4 * kk + u0;
        kband[kk].x = kb(K0 - l15);
        kband[kk].y = kb(K0 + 1 - l15);
    }

    // raw (un-mixed) field value of this wave's input channel at patch (q,col)
    auto rawVal = [&](int f, int q, int col, bool valid) -> float {
        int cl = (col < 28) ? col : 27;      // clamp: masked below
        float x1 = sP1[c][q][cl];
        float x2 = sP2[c][q][cl];
        float v;
        switch (f) {
            case 0:  v = x1;      break;
            case 1:  v = x2;      break;
            case 2:  v = x1 * x1; break;
            case 3:  v = x2 * x2; break;
            default: v = x1 * x2; break;
        }
        return valid ? v : 0.0f;
    };

    // ---- Phase A: blur 5 raw fields of input channel c ----
#pragma unroll
    for (int f = 0; f < 5; ++f) {
        // pass 1: blur along H.  T(16x28) = Kband(16x28) x P(28x28)
#pragma unroll
        for (int j = 0; j < 2; ++j) {
            v8f acc = {};
            const int col   = 16 * j + l15;
            const bool valid = (col < 28);
#pragma unroll
            for (int kk = 0; kk < 7; ++kk) {
                int q0 = 4 * kk + u0;
                v2f bb;
                bb.x = rawVal(f, q0, col, valid);
                bb.y = rawVal(f, q0 + 1, col, valid);
                acc = __builtin_amdgcn_wmma_f32_16x16x4_f32(
                    false, kband[kk], false, bb, (short)0, acc, false, false);
            }
#pragma unroll
            for (int v = 0; v < 8; ++v)
                sT[c][v + 8 * half][16 * j + l15] = acc[v];  // D layout -> LDS
        }
        __syncthreads();

        // pass 2: blur along W.  Out(16x16) = T(16x28) x Wband(28x16)
        v8f acc2 = {};
#pragma unroll
        for (int kk = 0; kk < 7; ++kk) {
            int base = 4 * kk + u0;
            v2f aa;                       // A layout from staged T
            aa.x = sT[c][l15][base];
            aa.y = sT[c][l15][base + 1];
            acc2 = __builtin_amdgcn_wmma_f32_16x16x4_f32(
                false, aa, false, kband[kk], (short)0, acc2, false, false);
        }
#pragma unroll
        for (int v = 0; v < 8; ++v)
            sOut[c][f][lane * 8 + v] = acc2[v];
        __syncthreads();   // protects sT reuse; last one publishes sOut
    }

    // ---- Phase B: 3x3 channel mix for output channel c + SSIM + reduce ----
    const float m0 = sc.M[c][0], m1 = sc.M[c][1], m2 = sc.M[c][2];
    const float C1v = 1.0e-4f;   // (0.01*1.0)^2
    const float C2v = 9.0e-4f;   // (0.03*1.0)^2
    float lsum = 0.0f;
#pragma unroll
    for (int v = 0; v < 8; ++v) {
        const int e = lane * 8 + v;
        float Fv[5];
#pragma unroll
        for (int f = 0; f < 5; ++f)
            Fv[f] = m0 * sOut[0][f][e] + m1 * sOut[1][f][e] + m2 * sOut[2][f][e];
        float mu1 = Fv[0], mu2 = Fv[1];
        float mu1sq = mu1 * mu1, mu2sq = mu2 * mu2, mu12 = mu1 * mu2;
        float s1  = Fv[2] - mu1sq;
        float s2  = Fv[3] - mu2sq;
        float s12 = Fv[4] - mu12;
        float num = (2.0f * mu12 + C1v) * (2.0f * s12 + C2v);
        float den = (mu1sq + mu2sq + C1v) * (s1 + s2 + C2v);
        lsum += num / den;
    }
#pragma unroll
    for (int off = 16; off > 0; off >>= 1)
        lsum += __shfl_down(lsum, off, 32);
    if (lane == 0) sRed[c] = lsum;
    __syncthreads();
    if (tid == 0) partials[blockIdx.x] = sRed[0] + sRed[1] + sRed[2];
}

__global__ __launch_bounds__(1024) void ssim_reduce_kernel(
    const float* __restrict__ partials, int nparts, float* __restrict__ out, float inv)
{
    __shared__ float red[1024];
    float s = 0.0f;
    for (int i = threadIdx.x; i < nparts; i += 1024) s += partials[i];
    red[threadIdx.x] = s;
    __syncthreads();
    for (int off = 512; off > 0; off >>= 1) {
        if ((int)threadIdx.x < off) red[threadIdx.x] += red[threadIdx.x + off];
        __syncthreads();
    }
    if (threadIdx.x == 0) out[0] = red[0] * inv;
}

extern "C" void kernel_launch(void* const* d_in, const int* in_sizes, int n_in,
                              void* d_out, int out_size, void* d_ws, size_t ws_size,
                              hipStream_t stream) {
    (void)in_sizes; (void)n_in; (void)out_size; (void)ws_size;
    const float* img1 = (const float*)d_in[0];
    const float* img2 = (const float*)d_in[1];
    float* out      = (float*)d_out;
    float* partials = (float*)d_ws;

    // Host-side constants (deterministic, recomputed every call).
    SsimConsts sc;
    {
        const float sigma = 1.5f;
        float ksum = 0.0f;
        for (int t = 0; t < 13; ++t) {
            float x = (float)(t - 6);
            sc.k[t] = expf(-0.5f * (x / sigma) * (x / sigma));
            ksum += sc.k[t];
        }
        for (int t = 0; t < 13; ++t) sc.k[t] /= ksum;
        // channel-axis blur over length-3 axis with reflect -> 3x3 matrix
        for (int cc = 0; cc < 3; ++cc) {
            for (int j = 0; j < 3; ++j) sc.M[cc][j] = 0.0f;
            for (int t = 0; t < 13; ++t) {
                int i  = cc + t - 6;
                int j6 = ((i % 6) + 6) % 6;
                int j3 = (j6 >= 3) ? (5 - j6) : j6;
                sc.M[cc][j3] += sc.k[t];
            }
        }
    }

    const int nblocks = NIMG * TILES * TILES;   // 32768
    ssim_tile_kernel<<<nblocks, 96, 0, stream>>>(img1, img2, partials, sc);

    const float inv = 1.0f / (float)((size_t)NIMG * NCH * HW * HW);
    ssim_reduce_kernel<<<1, 1024, 0, stream>>>(partials, nblocks, out, inv);
}